// PriorityQueueV2_77103252897913
// MI455X (gfx1250) — compile-verified
//
#include <hip/hip_runtime.h>
#include <math.h>

#define BB 8
#define NN 1024
#define MM 1024
#define EE 128
#define FF 128
#define HH 8
#define OO 128
#define SMALL_EPS 1e-6f
#define NEG_BIG -1e9f

typedef float v2f __attribute__((ext_vector_type(2)));
typedef float v8f __attribute__((ext_vector_type(8)));

// ---------------- K0: zero total_pop ----------------
__global__ void k_zero(float* p, int n) {
    int i = blockIdx.x * blockDim.x + threadIdx.x;
    if (i < n) p[i] = 0.0f;
}

// ---------------- K1: zsum -> write_values (tanh), push (sigmoid) ----------------
// grid(B), block(F=128)
__global__ void k_sumz(const float* __restrict__ z,
                       const float* __restrict__ W_val, const float* __restrict__ b_val,
                       const float* __restrict__ W_push, const float* __restrict__ b_push,
                       float* __restrict__ wv, float* __restrict__ push) {
    int b = blockIdx.x, f = threadIdx.x;
    __shared__ float zs[FF];
    const float* zb = z + (size_t)b * NN * FF;
    float s = 0.0f;
    for (int n = 0; n < NN; ++n) s += zb[n * FF + f];
    zs[f] = s;
    __syncthreads();
    // write_values column e = f  (E == F == 128)
    float acc = 0.0f;
    for (int k = 0; k < FF; ++k) acc += zs[k] * W_val[k * EE + f];
    wv[b * EE + f] = tanhf(acc + (float)NN * b_val[f]);
    if (f == 0) {
        float p = 0.0f;
        for (int k = 0; k < FF; ++k) p += zs[k] * W_push[k];
        p += (float)NN * b_push[0];
        push[b] = 1.0f / (1.0f + __expf(-p));
    }
}

// ---------------- K2: pop = sum(logits) * softmax(logits) over N ----------------
// grid(B), block(N=1024)
__global__ void k_pop(const float* __restrict__ z, const float* __restrict__ W_pop,
                      const float* __restrict__ b_pop, float* __restrict__ pop) {
    int b = blockIdx.x, n = threadIdx.x;
    __shared__ float red[NN];
    const float* zr = z + ((size_t)b * NN + n) * FF;
    float l = 0.0f;
    for (int k = 0; k < FF; ++k) l += zr[k] * W_pop[k];
    l += b_pop[0];
    // total sum of logits
    red[n] = l; __syncthreads();
    for (int s = NN / 2; s > 0; s >>= 1) { if (n < s) red[n] += red[n + s]; __syncthreads(); }
    float tot = red[0]; __syncthreads();
    // max
    red[n] = l; __syncthreads();
    for (int s = NN / 2; s > 0; s >>= 1) { if (n < s) red[n] = fmaxf(red[n], red[n + s]); __syncthreads(); }
    float mx = red[0]; __syncthreads();
    // exp-sum
    float e = __expf(l - mx);
    red[n] = e; __syncthreads();
    for (int s = NN / 2; s > 0; s >>= 1) { if (n < s) red[n] += red[n + s]; __syncthreads(); }
    float es = red[0];
    pop[(size_t)b * NN + n] = tot * (e / es);
}

// ---------------- K3: att = X @ W + b  (rows x 128 @ 128 x 8) ----------------
__global__ void k_att(const float* __restrict__ X, const float* __restrict__ W,
                      const float* __restrict__ bvec, float* __restrict__ out, int rows) {
    int r = blockIdx.x * blockDim.x + threadIdx.x;
    if (r >= rows) return;
    const float* x = X + (size_t)r * FF;
    float acc[HH];
#pragma unroll
    for (int h = 0; h < HH; ++h) acc[h] = bvec[h];
    for (int k = 0; k < FF; ++k) {
        float xv = x[k];
#pragma unroll
        for (int h = 0; h < HH; ++h) acc[h] += xv * W[k * HH + h];
    }
#pragma unroll
    for (int h = 0; h < HH; ++h) out[(size_t)r * HH + h] = acc[h];
}

// ---------------- K4: attention softmax + H-combine + argmax + total_pop scatter ----------------
// grid(B * N/256), block(256); att2 tile + bias live in LDS
__global__ void k_main(const float* __restrict__ att1, const float* __restrict__ att2,
                       const float* __restrict__ rs, const float* __restrict__ pop,
                       const float* __restrict__ W_comb, const float* __restrict__ b_comb,
                       int* __restrict__ max_idx, float* __restrict__ total_pop) {
    int blk = blockIdx.x;
    int b = blk / (NN / 256);
    int n0 = (blk % (NN / 256)) * 256;
    __shared__ float a2[MM * HH];   // 32 KB
    __shared__ float bias[MM];      // 4 KB
    __shared__ float wc[HH];
    for (int i = threadIdx.x; i < MM * HH; i += 256) a2[i] = att2[(size_t)b * MM * HH + i];
    for (int i = threadIdx.x; i < MM; i += 256)
        bias[i] = (rs[(size_t)b * MM + i] < SMALL_EPS) ? NEG_BIG : 0.0f;
    if (threadIdx.x < HH) wc[threadIdx.x] = W_comb[threadIdx.x];
    __syncthreads();

    int n = n0 + (int)threadIdx.x;
    float a1[HH];
#pragma unroll
    for (int h = 0; h < HH; ++h) a1[h] = att1[((size_t)b * NN + n) * HH + h];

    // pass 1: per-h max of lrelu(a1+a2)+bias
    float mx[HH];
#pragma unroll
    for (int h = 0; h < HH; ++h) mx[h] = -3.4e38f;
    for (int m = 0; m < MM; ++m) {
        float bm = bias[m];
#pragma unroll
        for (int h = 0; h < HH; ++h) {
            float x = a1[h] + a2[m * HH + h];
            x = (x > 0.0f) ? x : 0.01f * x;
            x += bm;
            mx[h] = fmaxf(mx[h], x);
        }
    }
    // pass 2: per-h exp-sums
    float sm[HH];
#pragma unroll
    for (int h = 0; h < HH; ++h) sm[h] = 0.0f;
    for (int m = 0; m < MM; ++m) {
        float bm = bias[m];
#pragma unroll
        for (int h = 0; h < HH; ++h) {
            float x = a1[h] + a2[m * HH + h];
            x = (x > 0.0f) ? x : 0.01f * x;
            x += bm;
            sm[h] += __expf(x - mx[h]);
        }
    }
    float smi[HH];
#pragma unroll
    for (int h = 0; h < HH; ++h) smi[h] = 1.0f / sm[h];
    // pass 3: comb over H, lrelu+bias, argmax (softmax is monotonic -> argmax only)
    float bc = b_comb[0];
    float best = -3.4e38f;
    int bi = 0;
    for (int m = 0; m < MM; ++m) {
        float bm = bias[m];
        float comb = bc;
#pragma unroll
        for (int h = 0; h < HH; ++h) {
            float x = a1[h] + a2[m * HH + h];
            x = (x > 0.0f) ? x : 0.01f * x;
            x += bm;
            comb += __expf(x - mx[h]) * smi[h] * wc[h];
        }
        float y = ((comb > 0.0f) ? comb : 0.01f * comb) + bm;
        if (y > best) { best = y; bi = m; }
    }
    max_idx[(size_t)b * NN + n] = bi;
    atomicAdd(&total_pop[(size_t)b * MM + bi], pop[(size_t)b * NN + n]);
}

// ---------------- K5: pop_given, new_rs, new_mem, new_write_mask ----------------
// grid(B), block(M=1024)
__global__ void k_finalize(const float* __restrict__ rs, const float* __restrict__ wm,
                           const float* __restrict__ mem, const float* __restrict__ wv,
                           const float* __restrict__ push, const float* __restrict__ total_pop,
                           float* __restrict__ pop_given, float* __restrict__ out_newmem,
                           float* __restrict__ out_newrs, float* __restrict__ out_newwm) {
    int b = blockIdx.x, m = threadIdx.x;
    size_t bm = (size_t)b * MM + m;
    float tp = total_pop[bm];
    float r = rs[bm];
    float pg = fminf(r, tp);
    pop_given[bm] = pg;
    float nr = r - pg;
    nr = (nr < SMALL_EPS) ? 0.0f : nr;
    float w = wm[bm];
    nr += push[b] * w;
    out_newrs[bm] = nr;
    const float* mrow = mem + bm * EE;
    float* orow = out_newmem + bm * EE;
    for (int e = 0; e < EE; ++e) orow[e] = mrow[e] + wv[b * EE + e] * w;
    // next_place = first m with new_rs < SMALL (else 0)
    __shared__ int firsti;
    if (m == 0) firsti = 0x7FFFFFFF;
    __syncthreads();
    if (nr < SMALL_EPS) atomicMin(&firsti, m);
    __syncthreads();
    int np = (firsti == 0x7FFFFFFF) ? 0 : firsti;
    out_newwm[bm] = (m == np) ? 1.0f : 0.0f;
}

// ---------------- K6: gather scaled memory rows (one-hot einsum collapse) ----------------
__global__ void k_gather(const int* __restrict__ max_idx, const float* __restrict__ pop,
                         const float* __restrict__ total_pop, const float* __restrict__ pop_given,
                         const float* __restrict__ mem, float* __restrict__ G) {
    int r = blockIdx.x * blockDim.x + threadIdx.x;  // b*N + n
    if (r >= BB * NN) return;
    int b = r / NN;
    int mi = max_idx[r];
    size_t bm = (size_t)b * MM + mi;
    float s = pop[r] / (total_pop[bm] + SMALL_EPS) * pop_given[bm];
    const float* mrow = mem + bm * EE;
    float* g = G + (size_t)r * EE;
    for (int e = 0; e < EE; ++e) g[e] = s * mrow[e];
}

// ---------------- K7: WMMA f32 GEMM  out = G(8192x128) @ W_out(128x128) + b_out ----------------
// grid(B*N/16 = 512), block(256) = 8 waves; wave w owns column-tile w; 32x v_wmma_f32_16x16x4_f32
__global__ void k_gemm(const float* __restrict__ G, const float* __restrict__ Wm,
                       const float* __restrict__ bout, float* __restrict__ out) {
    __shared__ float Gs[16 * FF];   // 8 KB A-strip
    int rowBase = blockIdx.x * 16;
    for (int i = threadIdx.x; i < 16 * FF; i += 256) Gs[i] = G[(size_t)rowBase * FF + i];
    __syncthreads();
    int wave = threadIdx.x >> 5;
    int lane = threadIdx.x & 31;
    int colBase = wave * 16;
    int lr = lane & 15;
    bool hi = lane >= 16;
    v8f acc = {};
    for (int k0 = 0; k0 < FF; k0 += 4) {
        int ka = k0 + (hi ? 2 : 0);
        // A 16x4 f32: lanes 0-15 hold K={0,1}, lanes 16-31 hold K={2,3} (ISA 7.12.2)
        v2f a = { Gs[lr * FF + ka], Gs[lr * FF + ka + 1] };
        // B 4x16 f32: row striped across lanes within a VGPR
        v2f bb = { Wm[ka * OO + colBase + lr], Wm[(ka + 1) * OO + colBase + lr] };
        acc = __builtin_amdgcn_wmma_f32_16x16x4_f32(
            false, a, false, bb, (short)0, acc, false, false);
    }
#pragma unroll
    for (int rr = 0; rr < 8; ++rr) {
        int row = rowBase + (hi ? rr + 8 : rr);
        int col = colBase + lr;
        out[(size_t)row * OO + col] = acc[rr] + bout[col];
    }
}

extern "C" void kernel_launch(void* const* d_in, const int* in_sizes, int n_in,
                              void* d_out, int out_size, void* d_ws, size_t ws_size,
                              hipStream_t stream) {
    const float* z      = (const float*)d_in[0];
    const float* mem    = (const float*)d_in[1];
    const float* rs     = (const float*)d_in[2];
    const float* wm     = (const float*)d_in[3];
    const float* W_push = (const float*)d_in[4];
    const float* b_push = (const float*)d_in[5];
    const float* W_pop  = (const float*)d_in[6];
    const float* b_pop  = (const float*)d_in[7];
    // W_val/b_val: 8,9 ; W_out/b_out: 10,11 ; W_a1/b_a1: 12,13 ; W_a2/b_a2: 14,15 ; W_comb/b_comb: 16,17
    const float* W_val  = (const float*)d_in[8];
    const float* b_val  = (const float*)d_in[9];
    const float* W_out  = (const float*)d_in[10];
    const float* b_out  = (const float*)d_in[11];
    const float* W_a1   = (const float*)d_in[12];
    const float* b_a1   = (const float*)d_in[13];
    const float* W_a2   = (const float*)d_in[14];
    const float* b_a2   = (const float*)d_in[15];
    const float* W_comb = (const float*)d_in[16];
    const float* b_comb = (const float*)d_in[17];

    float* out        = (float*)d_out;
    float* out_output = out;                       // B*N*O
    float* out_newmem = out_output + BB * NN * OO; // B*M*E
    float* out_newrs  = out_newmem + BB * MM * EE; // B*M
    float* out_newwm  = out_newrs + BB * MM;       // B*M

    float* wsf       = (float*)d_ws;
    float* wv        = wsf;                          // B*E
    float* push      = wv + BB * EE;                 // B
    float* pop       = push + BB;                    // B*N
    float* att1      = pop + BB * NN;                // B*N*H
    float* att2      = att1 + (size_t)BB * NN * HH;  // B*M*H
    float* total_pop = att2 + (size_t)BB * MM * HH;  // B*M
    float* pop_given = total_pop + BB * MM;          // B*M
    float* G         = pop_given + BB * MM;          // B*N*E
    int*   max_idx   = (int*)(G + (size_t)BB * NN * EE); // B*N

    k_zero<<<(BB * MM + 255) / 256, 256, 0, stream>>>(total_pop, BB * MM);
    k_sumz<<<BB, FF, 0, stream>>>(z, W_val, b_val, W_push, b_push, wv, push);
    k_pop<<<BB, NN, 0, stream>>>(z, W_pop, b_pop, pop);
    k_att<<<(BB * NN + 255) / 256, 256, 0, stream>>>(z, W_a1, b_a1, att1, BB * NN);
    k_att<<<(BB * MM + 255) / 256, 256, 0, stream>>>(mem, W_a2, b_a2, att2, BB * MM);
    k_main<<<BB * (NN / 256), 256, 0, stream>>>(att1, att2, rs, pop, W_comb, b_comb,
                                                max_idx, total_pop);
    k_finalize<<<BB, MM, 0, stream>>>(rs, wm, mem, wv, push, total_pop,
                                      pop_given, out_newmem, out_newrs, out_newwm);
    k_gather<<<(BB * NN + 255) / 256, 256, 0, stream>>>(max_idx, pop, total_pop, pop_given,
                                                        mem, G);
    k_gemm<<<BB * NN / 16, 256, 0, stream>>>(G, W_out, b_out, out_output);
}